// MPG_84464826843561
// MI455X (gfx1250) — compile-verified
//
#include <hip/hip_runtime.h>
#include <hip/hip_bf16.h>

typedef __attribute__((ext_vector_type(2))) float v2f;
typedef __attribute__((ext_vector_type(8))) float v8f;

#define FEAT 768
#define MB   128
#define NB   128
#define KC   32
#define LDA  34    // As row stride: even (8B-aligned b64 frags), conflict-free bank map
#define LDBP 288   // Bs pair-row stride (floats): 288 % 64 == 32 -> halves on disjoint banks

// ---------------- degree / normalization ----------------

__global__ __launch_bounds__(256) void MPG_deg_init(float* deg, int n) {
    int i = blockIdx.x * 256 + threadIdx.x;
    if (i < n) deg[i] = 1.0f;  // self-loop contributes 1
}

__global__ __launch_bounds__(256) void MPG_deg_accum(float* deg, const int* __restrict__ ei,
                                                     int E) {
    int e = blockIdx.x * 256 + threadIdx.x;
    if (e < E) atomicAdd(&deg[ei[E + e]], 1.0f);  // col = target
}

__global__ __launch_bounds__(256) void MPG_deg_rsqrt(float* deg, int n) {
    int i = blockIdx.x * 256 + threadIdx.x;
    if (i < n) {
        float d = deg[i];
        deg[i] = (d > 0.0f) ? __frsqrt_rn(d) : 0.0f;
    }
}

// ---------------- h = x @ W via V_WMMA_F32_16X16X4_F32 ----------------
// Block: 256 thr = 8 wave32. Tile: 128M x 128N, K staged in 32-chunks.
// Bs is K-pair interleaved: Bs[p][n][2] so each B fragment is one ds_load_b64.

__global__ __launch_bounds__(256) void MPG_gemm_wmma(const float* __restrict__ X,
                                                     const float* __restrict__ W,
                                                     float* __restrict__ H, int nNodes) {
    __shared__ float As[MB * LDA];            // [row][k]          17408 B
    __shared__ float Bs[(KC / 2) * LDBP];     // [kpair][n][2]     18432 B

    const int tid   = threadIdx.x;
    const int lane  = tid & 31;
    const int wave  = tid >> 5;     // 8 waves, each owns 16 M-rows
    const int half  = lane >> 4;    // 0: K pair 0/1 ; 1: K pair 2/3
    const int l16   = lane & 15;
    const int m_blk = blockIdx.x * MB;
    const int n_blk = blockIdx.y * NB;

    v8f acc0 = {}, acc1 = {}, acc2 = {}, acc3 = {};
    v8f acc4 = {}, acc5 = {}, acc6 = {}, acc7 = {};

    for (int k0 = 0; k0 < FEAT; k0 += KC) {
        // Stage As: 128 rows x 32 k = 1024 float4 slots, 4 per thread, coalesced on K.
#pragma unroll
        for (int j = 0; j < 4; ++j) {
            int slot = tid + 256 * j;
            int row  = slot >> 3;          // 0..127
            int kq   = (slot & 7) << 2;    // 0,4,..,28
            int grow = m_blk + row;
            float4 v = make_float4(0.f, 0.f, 0.f, 0.f);
            if (grow < nNodes)
                v = *(const float4*)(X + (size_t)grow * FEAT + k0 + kq);
            float* dst = &As[row * LDA + kq];   // 8B-aligned (row*34 + kq even)
            dst[0] = v.x; dst[1] = v.y; dst[2] = v.z; dst[3] = v.w;
        }
        // Stage Bs interleaved: 16 k-pairs x 32 n-quads = 512 slots, 2 per thread.
#pragma unroll
        for (int j = 0; j < 2; ++j) {
            int slot = tid + 256 * j;
            int p    = slot >> 5;           // 0..15 (k pair)
            int nq   = (slot & 31) << 2;    // 0,4,..,124
            const float* wp = W + (size_t)(k0 + 2 * p) * FEAT + n_blk + nq;
            float4 v0 = *(const float4*)(wp);
            float4 v1 = *(const float4*)(wp + FEAT);
            float* dst = &Bs[p * LDBP + nq * 2];   // 8B-aligned pairs
            dst[0] = v0.x; dst[1] = v1.x;
            dst[2] = v0.y; dst[3] = v1.y;
            dst[4] = v0.z; dst[5] = v1.z;
            dst[6] = v0.w; dst[7] = v1.w;
        }
        __syncthreads();

        const float* arow = &As[(wave * 16 + l16) * LDA];
#pragma unroll
        for (int kk = 0; kk < KC; kk += 4) {
            const int ka = kk + half * 2;            // even
            v2f a = *(const v2f*)(arow + ka);        // ds_load_b64
            const float* brow = &Bs[(ka >> 1) * LDBP + l16 * 2];
            v2f b0 = *(const v2f*)(brow + 0);        // 8 x ds_load_b64, bank-clean
            v2f b1 = *(const v2f*)(brow + 32);
            v2f b2 = *(const v2f*)(brow + 64);
            v2f b3 = *(const v2f*)(brow + 96);
            v2f b4 = *(const v2f*)(brow + 128);
            v2f b5 = *(const v2f*)(brow + 160);
            v2f b6 = *(const v2f*)(brow + 192);
            v2f b7 = *(const v2f*)(brow + 224);
            acc0 = __builtin_amdgcn_wmma_f32_16x16x4_f32(false, a, false, b0, (short)0, acc0, false, false);
            acc1 = __builtin_amdgcn_wmma_f32_16x16x4_f32(false, a, false, b1, (short)0, acc1, false, false);
            acc2 = __builtin_amdgcn_wmma_f32_16x16x4_f32(false, a, false, b2, (short)0, acc2, false, false);
            acc3 = __builtin_amdgcn_wmma_f32_16x16x4_f32(false, a, false, b3, (short)0, acc3, false, false);
            acc4 = __builtin_amdgcn_wmma_f32_16x16x4_f32(false, a, false, b4, (short)0, acc4, false, false);
            acc5 = __builtin_amdgcn_wmma_f32_16x16x4_f32(false, a, false, b5, (short)0, acc5, false, false);
            acc6 = __builtin_amdgcn_wmma_f32_16x16x4_f32(false, a, false, b6, (short)0, acc6, false, false);
            acc7 = __builtin_amdgcn_wmma_f32_16x16x4_f32(false, a, false, b7, (short)0, acc7, false, false);
        }
        __syncthreads();
    }

    // C/D layout: VGPR v -> M = v + half*8, N = l16 (+16 per tile)
#pragma unroll
    for (int v = 0; v < 8; ++v) {
        int grow = m_blk + wave * 16 + v + half * 8;
        if (grow < nNodes) {
            float* dst = H + (size_t)grow * FEAT + n_blk + l16;
            dst[0]   = acc0[v];
            dst[16]  = acc1[v];
            dst[32]  = acc2[v];
            dst[48]  = acc3[v];
            dst[64]  = acc4[v];
            dst[80]  = acc5[v];
            dst[96]  = acc6[v];
            dst[112] = acc7[v];
        }
    }
}

// ---------------- out = h * dinv^2 + b (self-loop + bias, clears d_out) ----------------

__global__ __launch_bounds__(256) void MPG_out_init(float* __restrict__ out,
                                                    const float* __restrict__ h,
                                                    const float* __restrict__ dinv,
                                                    const float* __restrict__ b, int n) {
    int idx = blockIdx.x * 256 + threadIdx.x;
    if (idx < n * FEAT) {
        int node = idx / FEAT;
        int f    = idx - node * FEAT;
        float di = dinv[node];
        out[idx] = h[idx] * di * di + b[f];
    }
}

// ---------------- edge scatter: one wave32 per edge, float4 gather + f32 atomics ----------------

__global__ __launch_bounds__(256) void MPG_scatter(float* __restrict__ out,
                                                   const float* __restrict__ h,
                                                   const float* __restrict__ dinv,
                                                   const int* __restrict__ ei, int E) {
    int wave = threadIdx.x >> 5;
    int lane = threadIdx.x & 31;
    int e    = blockIdx.x * 8 + wave;
    if (e >= E) return;
    int r = ei[e];        // source
    int c = ei[E + e];    // target
    float w = dinv[r] * dinv[c];
    const float* hr = h + (size_t)r * FEAT;
    float*       oc = out + (size_t)c * FEAT;
#pragma unroll
    for (int i = 0; i < FEAT / 128; ++i) {
        int f = i * 128 + lane * 4;
        float4 v = *(const float4*)(hr + f);  // coalesced, L2-resident
        atomicAdd(oc + f + 0, w * v.x);
        atomicAdd(oc + f + 1, w * v.y);
        atomicAdd(oc + f + 2, w * v.z);
        atomicAdd(oc + f + 3, w * v.w);
    }
}

extern "C" void kernel_launch(void* const* d_in, const int* in_sizes, int n_in,
                              void* d_out, int out_size, void* d_ws, size_t ws_size,
                              hipStream_t stream) {
    const float* x  = (const float*)d_in[0];
    const int*   ei = (const int*)d_in[1];
    const float* W  = (const float*)d_in[2];
    const float* b  = (const float*)d_in[3];
    float*       out = (float*)d_out;

    const int N = in_sizes[0] / FEAT;
    const int E = in_sizes[1] / 2;

    float* dinv = (float*)d_ws;                        // N floats
    float* h    = (float*)((char*)d_ws + 65536);       // N x 768 floats

    MPG_deg_init<<<(N + 255) / 256, 256, 0, stream>>>(dinv, N);
    MPG_deg_accum<<<(E + 255) / 256, 256, 0, stream>>>(dinv, ei, E);
    MPG_deg_rsqrt<<<(N + 255) / 256, 256, 0, stream>>>(dinv, N);

    dim3 grid((N + MB - 1) / MB, FEAT / NB);
    MPG_gemm_wmma<<<grid, 256, 0, stream>>>(x, W, h, N);

    MPG_out_init<<<(N * FEAT + 255) / 256, 256, 0, stream>>>(out, h, dinv, b, N);
    MPG_scatter<<<(E + 7) / 8, 256, 0, stream>>>(out, h, dinv, ei, E);
}